// GNNClassifier_22067541967321
// MI455X (gfx1250) — compile-verified
//
#include <hip/hip_runtime.h>
#include <hip/hip_bf16.h>

// ---------------------------------------------------------------------------
// GCN (2x GCNConv + mean-pool + MLP head), fp32 end-to-end.
// GEMMs use native fp32 WMMA (V_WMMA_F32_16X16X4_F32); W staged to LDS via
// the Tensor Data Mover (tensor_load_to_lds + s_wait_tensorcnt).
// ---------------------------------------------------------------------------

typedef __attribute__((ext_vector_type(2))) float        v2f;
typedef __attribute__((ext_vector_type(8))) float        v8f;
typedef __attribute__((ext_vector_type(4))) unsigned int u32x4;
typedef __attribute__((ext_vector_type(8))) int          i32x8;
typedef __attribute__((ext_vector_type(4))) int          i32x4;

#define N_FEAT 128   // IN_DIM == HID == 128
#define NGRAPH 64
#define HALF_HID 64
#define OUT_DIM 8

#if __has_builtin(__builtin_amdgcn_tensor_load_to_lds) && __has_builtin(__builtin_amdgcn_s_wait_tensorcnt)
#define USE_TDM 1
#else
#define USE_TDM 0
#endif

// ---------------------------------------------------------------------------
// norm computation: deg = 1 (self loop) + in-degree; norm = rsqrt(deg)
// ---------------------------------------------------------------------------
__global__ void k_fill1(float* p, int n) {
    int i = blockIdx.x * blockDim.x + threadIdx.x;
    if (i < n) p[i] = 1.0f;
}

__global__ void k_deg(const int* __restrict__ dst, float* __restrict__ deg, int E) {
    int e = blockIdx.x * blockDim.x + threadIdx.x;
    if (e < E) atomicAdd(&deg[dst[e]], 1.0f);
}

__global__ void k_rsqrt(float* p, int n) {
    int i = blockIdx.x * blockDim.x + threadIdx.x;
    if (i < n) p[i] = __frsqrt_rn(fmaxf(p[i], 1.0f));
}

// ---------------------------------------------------------------------------
// Pack W[128,128] into pair-interleaved layout so a WMMA B fragment
// (K rows k, k+1 at column n) is one contiguous 8-byte LDS load:
//   Wp[(k>>1)*256 + n*2 + (k&1)] = W[k*128 + n]
// ---------------------------------------------------------------------------
__global__ void k_pack_w(const float* __restrict__ W, float* __restrict__ Wp) {
    int i = blockIdx.x * blockDim.x + threadIdx.x;
    if (i < N_FEAT * N_FEAT) {
        int k = i >> 7, n = i & 127;
        Wp[((k >> 1) << 8) + (n << 1) + (k & 1)] = W[i];
    }
}

// ---------------------------------------------------------------------------
// Tensor Data Mover: async copy of the 64KB packed W (16384 fp32, 1-D tile)
// from global memory into LDS. D# built per ISA 8.3/8.4 field layout.
// ---------------------------------------------------------------------------
__device__ __forceinline__ void tdm_load_w(const float* __restrict__ gsrc,
                                           float* __restrict__ lds_dst) {
#if USE_TDM
    unsigned long long ga = (unsigned long long)(__SIZE_TYPE__)gsrc;
    unsigned int lds = (unsigned int)(__SIZE_TYPE__)lds_dst;  // flat addr low 32 = LDS offset

    u32x4 g0;
    g0[0] = 1u;                                      // count=1, user-mode descriptor
    g0[1] = lds;                                     // lds_addr [63:32]
    g0[2] = (unsigned int)ga;                        // global_addr [95:64]
    g0[3] = ((unsigned int)(ga >> 32) & 0x01FFFFFFu) // global_addr [120:96]
            | (2u << 30);                            // type=2 ("image") [127:126]

    i32x8 g1;
    g1[0] = 0x20000;               // wg_mask=0, data_size=2 (4B)
    g1[1] = (int)(16384u << 16);   // tensor_dim0[15:0]=16384 in [63:48]
    g1[2] = (int)(1u << 16);       // tensor_dim0 hi=0 | tensor_dim1=1
    g1[3] = (int)(16384u << 16);   // tensor_dim1 hi=0 | tile_dim0=16384
    g1[4] = 1;                     // tile_dim1=1, tile_dim2=0
    g1[5] = 16384;                 // tensor_dim0_stride[31:0]
    g1[6] = 0;                     // stride hi / tensor_dim1_stride lo
    g1[7] = 0;

    i32x4 z4 = {0, 0, 0, 0};
#if __has_include(<hip/amd_detail/amd_gfx1250_TDM.h>)
    i32x8 z8 = {0, 0, 0, 0, 0, 0, 0, 0};
    __builtin_amdgcn_tensor_load_to_lds(g0, g1, z4, z4, z8, 0);   // clang-23 6-arg form
#else
    __builtin_amdgcn_tensor_load_to_lds(g0, g1, z4, z4, 0);       // ROCm 7.2 5-arg form
#endif
#endif
}

// ---------------------------------------------------------------------------
// H[M,128] = act(X)[M,128] @ W[128,128]   (act = identity or relu, templated)
// Block: 256 thr = 8 waves; each wave computes a 32-row x 128-col strip
// (2 M-subtiles x 8 N-tiles = 16 fp32 accumulators, 128 VGPRs).
// Per K-step: 2 global b64 A loads + 8 ds_load_b64 B fragments feed 16 WMMAs.
// ---------------------------------------------------------------------------
template <int RELU>
__global__ __launch_bounds__(256) void k_gemm128(const float* __restrict__ X,
                                                 const float* __restrict__ Wp,
                                                 float* __restrict__ H, int M) {
    __shared__ float sW[N_FEAT * N_FEAT];  // 64 KB, pair-interleaved layout

#if USE_TDM
    if (threadIdx.x < 32) tdm_load_w(Wp, sW);           // one wave drives the TDM
    __builtin_amdgcn_s_wait_tensorcnt(0);
#else
    for (int i = threadIdx.x; i < (N_FEAT * N_FEAT) / 4; i += blockDim.x)
        ((float4*)sW)[i] = ((const float4*)Wp)[i];
#endif
    __syncthreads();

    const int wave = threadIdx.x >> 5;    // 0..7 (wave32)
    const int lane = threadIdx.x & 31;
    const int ncol = lane & 15;
    const int koff = (lane >> 4) * 2;     // A frag: lanes>=16 hold K=k0+2,k0+3
    const int rowbase = blockIdx.x * 256 + wave * 32;

    // Clamp OOB rows to M-1: garbage only feeds output rows that are never
    // stored, so no per-iteration masking is needed in the K loop.
    const float2* __restrict__ A2[2];
#pragma unroll
    for (int s2 = 0; s2 < 2; ++s2) {
        int mr = rowbase + s2 * 16 + (lane & 15);
        if (mr >= M) mr = M - 1;
        A2[s2] = (const float2*)(X + (size_t)mr * N_FEAT);
    }

    v8f acc[2][8];
#pragma unroll
    for (int s2 = 0; s2 < 2; ++s2)
#pragma unroll
        for (int j = 0; j < 8; ++j)
            acc[s2][j] = v8f{0.f, 0.f, 0.f, 0.f, 0.f, 0.f, 0.f, 0.f};

    const v2f* sWp = (const v2f*)sW;

    for (int k0 = 0; k0 < N_FEAT; k0 += 4) {
        v2f a[2];
#pragma unroll
        for (int s2 = 0; s2 < 2; ++s2) {
            float2 av = A2[s2][(k0 + koff) >> 1];  // K = k0+koff, k0+koff+1
            a[s2].x = av.x;
            a[s2].y = av.y;
            if (RELU) {
                a[s2].x = fmaxf(a[s2].x, 0.f);
                a[s2].y = fmaxf(a[s2].y, 0.f);
            }
        }
        const v2f* __restrict__ brow = sWp + ((k0 + koff) >> 1) * N_FEAT;
#pragma unroll
        for (int j = 0; j < 8; ++j) {
            v2f b = brow[j * 16 + ncol];  // single ds_load_b64, even reg pair
            acc[0][j] = __builtin_amdgcn_wmma_f32_16x16x4_f32(
                false, a[0], false, b, (short)0, acc[0][j], false, false);
            acc[1][j] = __builtin_amdgcn_wmma_f32_16x16x4_f32(
                false, a[1], false, b, (short)0, acc[1][j], false, false);
        }
    }

    // C/D layout: VGPR r -> row r + 8*(lane>=16), col = lane&15
    const int rb = (lane >> 4) * 8;
#pragma unroll
    for (int s2 = 0; s2 < 2; ++s2)
#pragma unroll
        for (int j = 0; j < 8; ++j)
#pragma unroll
            for (int r = 0; r < 8; ++r) {
                int m = rowbase + s2 * 16 + rb + r;
                if (m < M) H[(size_t)m * N_FEAT + j * 16 + ncol] = acc[s2][j][r];
            }
}

// ---------------------------------------------------------------------------
// Self-loop + bias init: Hout[n,:] = norm[n]^2 * Hin[n,:] + b[:]
// ---------------------------------------------------------------------------
__global__ void k_self_bias(const float* __restrict__ norm,
                            const float* __restrict__ Hin,
                            const float* __restrict__ b,
                            float* __restrict__ Hout, int Nn) {
    int t = blockIdx.x * blockDim.x + threadIdx.x;
    int n = t >> 5, lane = t & 31;
    if (n >= Nn) return;
    float c = norm[n] * norm[n];
    float4 v  = ((const float4*)(Hin + (size_t)n * N_FEAT))[lane];
    float4 bb = ((const float4*)b)[lane];
    float4 o = {v.x * c + bb.x, v.y * c + bb.y, v.z * c + bb.z, v.w * c + bb.w};
    ((float4*)(Hout + (size_t)n * N_FEAT))[lane] = o;
}

// ---------------------------------------------------------------------------
// Edge aggregation: Hout[d,:] += norm[s]*norm[d] * Hin[s,:]
// 32 lanes per edge, float4 gather + 4 fp32 atomics (L2-resident working set).
// ---------------------------------------------------------------------------
__global__ void k_edge_agg(const int* __restrict__ src, const int* __restrict__ dst,
                           const float* __restrict__ norm,
                           const float* __restrict__ Hin,
                           float* __restrict__ Hout, int E) {
    int t = blockIdx.x * blockDim.x + threadIdx.x;
    int e = t >> 5, lane = t & 31;
    if (e >= E) return;

    // warm L2 for a future edge's gather row (global_prefetch_b8)
    int ep = e + 2048; if (ep >= E) ep = E - 1;
    __builtin_prefetch(Hin + (size_t)src[ep] * N_FEAT + lane * 4, 0, 1);

    int s = src[e], d = dst[e];
    float c = norm[s] * norm[d];
    float4 v = ((const float4*)(Hin + (size_t)s * N_FEAT))[lane];
    float* out = Hout + (size_t)d * N_FEAT + lane * 4;
    atomicAdd(out + 0, v.x * c);
    atomicAdd(out + 1, v.y * c);
    atomicAdd(out + 2, v.z * c);
    atomicAdd(out + 3, v.w * c);
}

// ---------------------------------------------------------------------------
// Pooling: sums[g,:] += relu(H[n,:]); counts[g] += 1   (relu fused on load)
// ---------------------------------------------------------------------------
__global__ void k_zero(float* p, int n) {
    int i = blockIdx.x * blockDim.x + threadIdx.x;
    if (i < n) p[i] = 0.0f;
}

__global__ void k_pool(const float* __restrict__ H, const int* __restrict__ batch,
                       float* __restrict__ sums, float* __restrict__ counts, int Nn) {
    int t = blockIdx.x * blockDim.x + threadIdx.x;
    int n = t >> 5, lane = t & 31;
    if (n >= Nn) return;
    int g = batch[n];
    float4 v = ((const float4*)(H + (size_t)n * N_FEAT))[lane];
    v.x = fmaxf(v.x, 0.f); v.y = fmaxf(v.y, 0.f);
    v.z = fmaxf(v.z, 0.f); v.w = fmaxf(v.w, 0.f);
    float* s = sums + g * N_FEAT + lane * 4;
    atomicAdd(s + 0, v.x);
    atomicAdd(s + 1, v.y);
    atomicAdd(s + 2, v.z);
    atomicAdd(s + 3, v.w);
    if (lane == 0) atomicAdd(&counts[g], 1.0f);
}

// ---------------------------------------------------------------------------
// Head: pooled = sums/counts; z = relu(pooled@Wc1 + bc1); out = z@Wc2 + bc2
// ---------------------------------------------------------------------------
__global__ __launch_bounds__(256) void k_head(const float* __restrict__ sums,
                                              const float* __restrict__ counts,
                                              const float* __restrict__ Wc1,
                                              const float* __restrict__ bc1,
                                              const float* __restrict__ Wc2,
                                              const float* __restrict__ bc2,
                                              float* __restrict__ out) {
    __shared__ float sP[NGRAPH * N_FEAT];   // 32 KB
    __shared__ float sZ[NGRAPH * HALF_HID]; // 16 KB
    int t = threadIdx.x;

    for (int i = t; i < NGRAPH * N_FEAT; i += 256) {
        int g = i >> 7;
        sP[i] = sums[i] / fmaxf(counts[g], 1.0f);
    }
    __syncthreads();

    for (int i = t; i < NGRAPH * HALF_HID; i += 256) {
        int g = i >> 6, c = i & 63;
        float acc = bc1[c];
        const float* pr = &sP[g * N_FEAT];
#pragma unroll 4
        for (int k = 0; k < N_FEAT; ++k) acc = fmaf(pr[k], Wc1[k * HALF_HID + c], acc);
        sZ[i] = fmaxf(acc, 0.f);
    }
    __syncthreads();

    for (int i = t; i < NGRAPH * OUT_DIM; i += 256) {
        int g = i >> 3, c = i & 7;
        float acc = bc2[c];
        const float* zr = &sZ[g * HALF_HID];
#pragma unroll 4
        for (int k = 0; k < HALF_HID; ++k) acc = fmaf(zr[k], Wc2[k * OUT_DIM + c], acc);
        out[i] = acc;
    }
}

// ---------------------------------------------------------------------------
// Host-side orchestration
// ---------------------------------------------------------------------------
extern "C" void kernel_launch(void* const* d_in, const int* in_sizes, int n_in,
                              void* d_out, int out_size, void* d_ws, size_t ws_size,
                              hipStream_t stream) {
    const float* x     = (const float*)d_in[0];
    const int*   ei    = (const int*)d_in[1];
    const int*   batch = (const int*)d_in[2];
    // d_in[3] = num_graphs scalar (compile-time 64)
    const float* W1  = (const float*)d_in[4];
    const float* b1  = (const float*)d_in[5];
    const float* W2  = (const float*)d_in[6];
    const float* b2  = (const float*)d_in[7];
    const float* Wc1 = (const float*)d_in[8];
    const float* bc1 = (const float*)d_in[9];
    const float* Wc2 = (const float*)d_in[10];
    const float* bc2 = (const float*)d_in[11];
    float* outp = (float*)d_out;

    const int N = in_sizes[0] / N_FEAT;   // 100000
    const int E = in_sizes[1] / 2;        // 1600000
    const int* srcp = ei;
    const int* dstp = ei + E;

    // workspace carve-up (fp32, 256B-aligned base from hipMalloc)
    float* norm = (float*)d_ws;
    size_t normElems = ((size_t)N + 63) & ~(size_t)63;
    float* wpack1 = norm + normElems;                  // [64,256] packed W1
    float* wpack2 = wpack1 + N_FEAT * N_FEAT;          // [64,256] packed W2
    float* buf1   = wpack2 + N_FEAT * N_FEAT;          // [N,128] X@W result
    float* buf2   = buf1 + (size_t)N * N_FEAT;         // [N,128] aggregated
    float* sums   = buf2 + (size_t)N * N_FEAT;         // [64,128]
    float* counts = sums + NGRAPH * N_FEAT;            // [64]

    const int TPB = 256;
    dim3 blk(TPB);
    int gN    = (N + TPB - 1) / TPB;
    int gE    = (E + TPB - 1) / TPB;
    int gN32  = (int)(((size_t)N * 32 + TPB - 1) / TPB);
    int gE32  = (int)(((size_t)E * 32 + TPB - 1) / TPB);
    int gGemm = (N + 255) / 256;
    int gPack = (N_FEAT * N_FEAT + TPB - 1) / TPB;

    // norms + weight packing
    k_fill1<<<gN, blk, 0, stream>>>(norm, N);
    k_deg<<<gE, blk, 0, stream>>>(dstp, norm, E);
    k_rsqrt<<<gN, blk, 0, stream>>>(norm, N);
    k_pack_w<<<gPack, blk, 0, stream>>>(W1, wpack1);
    k_pack_w<<<gPack, blk, 0, stream>>>(W2, wpack2);

    // layer 1: buf1 = X@W1 ; buf2 = D^-1/2 (A+I) D^-1/2 buf1 + b1
    k_gemm128<0><<<gGemm, blk, 0, stream>>>(x, wpack1, buf1, N);
    k_self_bias<<<gN32, blk, 0, stream>>>(norm, buf1, b1, buf2, N);
    k_edge_agg<<<gE32, blk, 0, stream>>>(srcp, dstp, norm, buf1, buf2, E);

    // layer 2: buf1 = relu(buf2)@W2 ; buf2 = agg(buf1) + b2  (relu fused on load)
    k_gemm128<1><<<gGemm, blk, 0, stream>>>(buf2, wpack2, buf1, N);
    k_self_bias<<<gN32, blk, 0, stream>>>(norm, buf1, b2, buf2, N);
    k_edge_agg<<<gE32, blk, 0, stream>>>(srcp, dstp, norm, buf1, buf2, E);

    // pooling (relu fused on load) + head
    k_zero<<<(NGRAPH * N_FEAT + NGRAPH + TPB - 1) / TPB, blk, 0, stream>>>(sums, NGRAPH * N_FEAT + NGRAPH);
    k_pool<<<gN32, blk, 0, stream>>>(buf2, batch, sums, counts, N);
    k_head<<<1, blk, 0, stream>>>(sums, counts, Wc1, bc1, Wc2, bc2, outp);
}